// LinearLoRAClassifier_36747740184729
// MI455X (gfx1250) — compile-verified
//
#include <hip/hip_runtime.h>
#include <hip/hip_bf16.h>

#define IN_FEATURES 2048
#define N_CLASSES   10000
#define BATCH       8192

typedef __attribute__((ext_vector_type(2))) float v2f;
typedef __attribute__((ext_vector_type(8))) float v8f;

// ---------------------------------------------------------------------------
// Phase 1: s[i] = dot(x[i,:], B[0,:])  (8192 dots of length 2048)
// One wave32 per row, 8 waves per block. float4 loads -> global_load_b128.
// ---------------------------------------------------------------------------
__global__ __launch_bounds__(256) void lora_rowdot_kernel(
    const float* __restrict__ x, const float* __restrict__ Bv,
    float* __restrict__ s) {
  const int lane = threadIdx.x & 31;
  const int wave = threadIdx.x >> 5;
  const int row  = blockIdx.x * 8 + wave;   // grid = 1024 blocks -> 8192 rows

  const float4* __restrict__ x4 = reinterpret_cast<const float4*>(
      x + (size_t)row * IN_FEATURES);
  const float4* __restrict__ b4 = reinterpret_cast<const float4*>(Bv);

  float acc = 0.0f;
#pragma unroll
  for (int it = 0; it < IN_FEATURES / 4 / 32; ++it) {   // 16 iters
    const int c = lane + it * 32;
    const float4 xv = x4[c];
    const float4 bv = b4[c];
    acc = fmaf(xv.x, bv.x, acc);
    acc = fmaf(xv.y, bv.y, acc);
    acc = fmaf(xv.z, bv.z, acc);
    acc = fmaf(xv.w, bv.w, acc);
  }

  // wave32 butterfly reduction
#pragma unroll
  for (int off = 16; off > 0; off >>= 1)
    acc += __shfl_xor(acc, off, 32);

  if (lane == 0) s[row] = acc;
}

// ---------------------------------------------------------------------------
// Phase 2: out[i,j] = relu(s[i] * A[j] + bias[j])
// One wave32 per 16x16 tile, computed by one V_WMMA_F32_16X16X4_F32 with
// bias pre-loaded into the C accumulator. 8192 = 512*16, 10000 = 625*16,
// so no bounds checks -> EXEC stays all-ones (WMMA requirement).
//
// A-matrix (16x4): VGPR0 = {K=0 | K=2}, VGPR1 = {K=1 | K=3}; only K=0 used.
// B-matrix (4x16): K=0 row in first half-VGPR; all other K slots zeroed.
// C/D (16x16):     VGPR r -> M = r (+8 for lanes 16-31), N = lane & 15.
// ---------------------------------------------------------------------------
__global__ __launch_bounds__(256) void lora_outer_wmma_kernel(
    const float* __restrict__ s, const float* __restrict__ w,
    const float* __restrict__ bias, float* __restrict__ out) {
  const int lane  = threadIdx.x & 31;
  const int wave  = threadIdx.x >> 5;
  const int tileJ = blockIdx.x;                 // 0..624
  const int tileI = blockIdx.y * 8 + wave;      // 0..511
  const int i0    = tileI << 4;
  const int j0    = tileJ << 4;
  const int l16   = lane & 15;

  // Converged loads (no EXEC-masked memory ops before the WMMA), then select.
  float sval = s[i0 + l16];
  float wval = w[j0 + l16];
  const float bv = bias[j0 + l16];
  sval = (lane < 16) ? sval : 0.0f;
  wval = (lane < 16) ? wval : 0.0f;

  v2f a;  a.x = sval; a.y = 0.0f;   // K=0 column of A-matrix, rest zero
  v2f b;  b.x = wval; b.y = 0.0f;   // K=0 row of B-matrix, rest zero
  v8f c;
#pragma unroll
  for (int r = 0; r < 8; ++r) c[r] = bv;  // bias broadcast along M (per-column)

  // D = A x B + C : emits v_wmma_f32_16x16x4_f32
  v8f d = __builtin_amdgcn_wmma_f32_16x16x4_f32(
      /*neg_a=*/false, a, /*neg_b=*/false, b,
      /*c_mod=*/(short)0, c, /*reuse_a=*/false, /*reuse_b=*/false);

  const size_t col   = (size_t)(j0 + l16);
  const int    rbase = i0 + ((lane >> 4) << 3);
#pragma unroll
  for (int r = 0; r < 8; ++r) {
    const float v = fmaxf(d[r], 0.0f);   // relu -> single v_max_num_f32
    out[(size_t)(rbase + r) * N_CLASSES + col] = v;
  }
}

// ---------------------------------------------------------------------------
// Inputs (setup_inputs order): 0: x [8192,2048] f32, 1: A [10000,1] f32,
//                              2: B [1,2048] f32,    3: bias [10000] f32
// Output: [8192,10000] f32. d_ws: 8192 floats of scratch for s.
// ---------------------------------------------------------------------------
extern "C" void kernel_launch(void* const* d_in, const int* in_sizes, int n_in,
                              void* d_out, int out_size, void* d_ws, size_t ws_size,
                              hipStream_t stream) {
  const float* x    = (const float*)d_in[0];
  const float* Aw   = (const float*)d_in[1];   // [N_CLASSES] (rank-1 column)
  const float* Bv   = (const float*)d_in[2];   // [IN_FEATURES]
  const float* bias = (const float*)d_in[3];
  float*       out  = (float*)d_out;
  float*       s    = (float*)d_ws;            // 8192 floats

  // Phase 1: row dots
  lora_rowdot_kernel<<<dim3(BATCH / 8), dim3(256), 0, stream>>>(x, Bv, s);

  // Phase 2: WMMA outer-product + bias + relu, one wave per 16x16 tile
  // grid: x = 625 j-tiles, y = 512/8 i-tile groups (8 waves per block)
  lora_outer_wmma_kernel<<<dim3(N_CLASSES / 16, BATCH / 16 / 8), dim3(256), 0,
                           stream>>>(s, Aw, bias, out);
}